// CBAM_21217138442464
// MI455X (gfx1250) — compile-verified
//
#include <hip/hip_runtime.h>
#include <math.h>

typedef float v2f __attribute__((ext_vector_type(2)));
typedef float v8f __attribute__((ext_vector_type(8)));

#define CCH   512
#define HW    3136            // 56*56
#define BATCH 32
#define CR    32              // C/R

// ---------------- Kernel 1: per-(b,c) avg/max pool over HW (float4 loads) ----------------
// One wave32 per (b,c). Output packed V[64][512]: rows 0..31 avg, rows 32..63 max.
__global__ __launch_bounds__(256) void k_pool(const float* __restrict__ x,
                                              float* __restrict__ Vbuf) {
    const int lane = threadIdx.x & 31;
    const int wv   = threadIdx.x >> 5;
    const int pair = blockIdx.x * 8 + wv;      // 0 .. 16383
    const int b = pair >> 9;                   // /512
    const int c = pair & 511;

    const float4* px4 = reinterpret_cast<const float4*>(x + (size_t)pair * HW);
    float s = 0.0f;
    float m = -INFINITY;
    // 784 float4 per (b,c): 24 full wave iterations + 16-lane tail
#pragma unroll 4
    for (int i = 0; i < 24; ++i) {
        float4 v = px4[i * 32 + lane];
        s += (v.x + v.y) + (v.z + v.w);
        m = fmaxf(m, fmaxf(fmaxf(v.x, v.y), fmaxf(v.z, v.w)));
    }
    if (lane < 16) {
        float4 v = px4[768 + lane];
        s += (v.x + v.y) + (v.z + v.w);
        m = fmaxf(m, fmaxf(fmaxf(v.x, v.y), fmaxf(v.z, v.w)));
    }
    // wave32 butterfly reduction
    for (int off = 16; off > 0; off >>= 1) {
        s += __shfl_xor(s, off, 32);
        m = fmaxf(m, __shfl_xor(m, off, 32));
    }
    if (lane == 0) {
        Vbuf[(size_t)b * CCH + c]        = s * (1.0f / (float)HW); // avg row b
        Vbuf[(size_t)(b + 32) * CCH + c] = m;                      // max row b+32
    }
}

// ---------------- Kernel 2: MLP via WMMA f32 16x16x4 ----------------
// Single workgroup, 8 waves (256 threads).
// GEMM1: H[64,32] = relu(V[64,512] @ w1[512,32])   (to LDS)
// GEMM2: Y[64,512] = H @ w2[32,512]; att[b,c] = sigmoid(Y[b,c] + Y[b+32,c])
__global__ __launch_bounds__(256) void k_mlp(const float* __restrict__ Vbuf,
                                             const float* __restrict__ w1,
                                             const float* __restrict__ w2,
                                             float* __restrict__ att) {
    __shared__ float H[64 * CR];   // 8 KB

    const int lane = threadIdx.x & 31;
    const int wv   = threadIdx.x >> 5;        // 0..7
    const int l16  = lane & 15;
    const int hi   = (lane >= 16) ? 1 : 0;    // lane half selects K pairs
    const int k0   = hi ? 2 : 0;              // VGPR0 K offset
    const int k1   = hi ? 3 : 1;              // VGPR1 K offset

    // ---- GEMM1: 4 M-tiles x 2 N-tiles, one tile per wave ----
    {
        const int mt = wv >> 1;               // 0..3
        const int nt = wv & 1;                // 0..1
        const int row = mt * 16 + l16;        // A: M striped on lanes 0-15 (mirrored 16-31)
        const int col = nt * 16 + l16;        // B: N striped on lanes
        v8f acc = {};
        for (int kk = 0; kk < CCH; kk += 4) {
            v2f a, bfrag;
            a.x = Vbuf[(size_t)row * CCH + kk + k0];
            a.y = Vbuf[(size_t)row * CCH + kk + k1];
            bfrag.x = w1[(size_t)(kk + k0) * CR + col];
            bfrag.y = w1[(size_t)(kk + k1) * CR + col];
            acc = __builtin_amdgcn_wmma_f32_16x16x4_f32(
                false, a, false, bfrag, (short)0, acc, false, false);
        }
        // ReLU + store tile to LDS. D layout: VGPR j -> M = j (lanes<16) / j+8 (lanes>=16)
#pragma unroll
        for (int j = 0; j < 8; ++j) {
            float v = acc[j];
            v = v > 0.0f ? v : 0.0f;
            int mrow = mt * 16 + j + (hi ? 8 : 0);
            H[mrow * CR + nt * 16 + l16] = v;
        }
    }
    __syncthreads();

    // ---- GEMM2: pair the avg tile (mt2) with the max tile (mt2+2) so the
    //      sigmoid(sum) is a per-lane register op. 32 N-tiles / 4 -> 8 per wave.
    {
        const int mt2 = wv & 1;               // rows 0..31 = avg part
        const int ntb = wv >> 1;              // 0..3
        const int rowA = mt2 * 16 + l16;          // avg rows
        const int rowM = (mt2 + 2) * 16 + l16;    // max rows
        for (int t = 0; t < 8; ++t) {
            const int nt = ntb + 4 * t;       // 0..31
            v8f accA = {};
            v8f accM = {};
            for (int kk = 0; kk < CR; kk += 4) {
                v2f aA, aM, bfrag;
                aA.x = H[rowA * CR + kk + k0];
                aA.y = H[rowA * CR + kk + k1];
                aM.x = H[rowM * CR + kk + k0];
                aM.y = H[rowM * CR + kk + k1];
                bfrag.x = w2[(size_t)(kk + k0) * CCH + nt * 16 + l16];
                bfrag.y = w2[(size_t)(kk + k1) * CCH + nt * 16 + l16];
                accA = __builtin_amdgcn_wmma_f32_16x16x4_f32(
                    false, aA, false, bfrag, (short)0, accA, false, false);
                accM = __builtin_amdgcn_wmma_f32_16x16x4_f32(
                    false, aM, false, bfrag, (short)0, accM, false, false);
            }
#pragma unroll
            for (int j = 0; j < 8; ++j) {
                int bidx = mt2 * 16 + j + (hi ? 8 : 0);  // batch 0..31
                int cidx = nt * 16 + l16;                // channel
                float z = accA[j] + accM[j];
                att[(size_t)bidx * CCH + cidx] = 1.0f / (1.0f + expf(-z));
            }
        }
    }
}

// ---------------- Kernel 3: out = x*att, plus channel mean/max per pixel ----------------
// Block: 256 thr = 16 channel-lanes x 16 pixel-quads (float4). 64 pixels per block,
// 49 chunks per batch (49*64 = 3136). All big traffic is b128.
__global__ __launch_bounds__(256) void k_chan_apply(const float* __restrict__ x,
                                                    const float* __restrict__ att,
                                                    float* __restrict__ out,
                                                    float* __restrict__ cat_avg,
                                                    float* __restrict__ cat_max) {
    __shared__ float ssum[16][64];
    __shared__ float smax[16][64];

    const int b     = blockIdx.x / 49;
    const int chunk = blockIdx.x % 49;
    const int p0    = chunk * 64;
    const int pq    = threadIdx.x & 15;    // pixel quad 0..15 -> pixels pq*4..pq*4+3
    const int cc    = threadIdx.x >> 4;    // channel lane 0..15

    float4 sum = make_float4(0.f, 0.f, 0.f, 0.f);
    float4 mx  = make_float4(-INFINITY, -INFINITY, -INFINITY, -INFINITY);

    for (int c = cc; c < CCH; c += 16) {
        float a = att[(size_t)b * CCH + c];
        size_t idx = ((size_t)b * CCH + c) * HW + p0 + pq * 4;
        float4 v = *reinterpret_cast<const float4*>(x + idx);
        v.x *= a; v.y *= a; v.z *= a; v.w *= a;
        *reinterpret_cast<float4*>(out + idx) = v;
        sum.x += v.x; sum.y += v.y; sum.z += v.z; sum.w += v.w;
        mx.x = fmaxf(mx.x, v.x); mx.y = fmaxf(mx.y, v.y);
        mx.z = fmaxf(mx.z, v.z); mx.w = fmaxf(mx.w, v.w);
    }
    ssum[cc][pq * 4 + 0] = sum.x; ssum[cc][pq * 4 + 1] = sum.y;
    ssum[cc][pq * 4 + 2] = sum.z; ssum[cc][pq * 4 + 3] = sum.w;
    smax[cc][pq * 4 + 0] = mx.x;  smax[cc][pq * 4 + 1] = mx.y;
    smax[cc][pq * 4 + 2] = mx.z;  smax[cc][pq * 4 + 3] = mx.w;
    __syncthreads();
    if (threadIdx.x < 64) {
        int p = threadIdx.x;
        float s = 0.0f;
        float m = -INFINITY;
#pragma unroll
        for (int i = 0; i < 16; ++i) {
            s += ssum[i][p];
            m = fmaxf(m, smax[i][p]);
        }
        cat_avg[(size_t)b * HW + p0 + p] = s * (1.0f / (float)CCH);
        cat_max[(size_t)b * HW + p0 + p] = m;
    }
}

// ---------------- Kernel 4: 7x7 conv over [B,2,56,56] + sigmoid -> sa ----------------
__global__ __launch_bounds__(256) void k_spatial_conv(const float* __restrict__ cat_avg,
                                                      const float* __restrict__ cat_max,
                                                      const float* __restrict__ cw,
                                                      float* __restrict__ sa) {
    int t = blockIdx.x * blockDim.x + threadIdx.x;
    if (t >= BATCH * HW) return;
    const int b = t / HW;
    const int p = t % HW;
    const int h = p / 56;
    const int w = p % 56;

    float acc = 0.0f;
#pragma unroll
    for (int ky = 0; ky < 7; ++ky) {
        int hy = h + ky - 3;
        if (hy < 0 || hy >= 56) continue;
#pragma unroll
        for (int kx = 0; kx < 7; ++kx) {
            int wx = w + kx - 3;
            if (wx < 0 || wx >= 56) continue;
            int q = hy * 56 + wx;
            float wa = cw[ky * 7 + kx];        // conv_w[0][0][ky][kx]
            float wm = cw[49 + ky * 7 + kx];   // conv_w[0][1][ky][kx]
            acc += wa * cat_avg[(size_t)b * HW + q] + wm * cat_max[(size_t)b * HW + q];
        }
    }
    sa[(size_t)b * HW + p] = 1.0f / (1.0f + expf(-acc));
}

// ---------------- Kernel 5: out *= sa (vectorized, in place) ----------------
__global__ __launch_bounds__(256) void k_scale(float* __restrict__ out,
                                               const float* __restrict__ sa) {
    size_t t = (size_t)blockIdx.x * blockDim.x + threadIdx.x;  // one float4 each
    size_t base = t * 4;                                       // HW%4==0 -> same (b,c)
    size_t bc = base / HW;
    int    p  = (int)(base % HW);
    int    b  = (int)(bc / CCH);

    float4 v = *reinterpret_cast<const float4*>(out + base);
    float4 s = *reinterpret_cast<const float4*>(sa + (size_t)b * HW + p);
    v.x *= s.x; v.y *= s.y; v.z *= s.z; v.w *= s.w;
    *reinterpret_cast<float4*>(out + base) = v;
}

extern "C" void kernel_launch(void* const* d_in, const int* in_sizes, int n_in,
                              void* d_out, int out_size, void* d_ws, size_t ws_size,
                              hipStream_t stream) {
    const float* x  = (const float*)d_in[0];   // [32,512,56,56]
    const float* w1 = (const float*)d_in[1];   // [512,32]
    const float* w2 = (const float*)d_in[2];   // [32,512]
    const float* cw = (const float*)d_in[3];   // [1,2,7,7]
    float* out = (float*)d_out;

    // workspace layout (floats)
    float* ws      = (float*)d_ws;
    float* Vbuf    = ws;                        // 64*512   = 32768
    float* att     = Vbuf + 64 * CCH;           // 32*512   = 16384
    float* cat_avg = att + BATCH * CCH;         // 32*3136  = 100352
    float* cat_max = cat_avg + BATCH * HW;      // 32*3136
    float* sa      = cat_max + BATCH * HW;      // 32*3136

    // 1) channel pooling: 16384 waves = 2048 blocks x 8 waves
    k_pool<<<2048, 256, 0, stream>>>(x, Vbuf);

    // 2) MLP (WMMA) -> att
    k_mlp<<<1, 256, 0, stream>>>(Vbuf, w1, w2, att);

    // 3) channel-attention apply + spatial stats (b128 traffic)
    k_chan_apply<<<BATCH * 49, 256, 0, stream>>>(x, att, out, cat_avg, cat_max);

    // 4) 7x7 conv + sigmoid
    k_spatial_conv<<<(BATCH * HW + 255) / 256, 256, 0, stream>>>(cat_avg, cat_max, cw, sa);

    // 5) out *= sa   (51380224 / 4 = 12845056 float4's)
    k_scale<<<12845056 / 256, 256, 0, stream>>>(out, sa);
}